// Router_64501818851344
// MI455X (gfx1250) — compile-verified
//
#include <hip/hip_runtime.h>
#include <stdint.h>

typedef __attribute__((ext_vector_type(2))) float v2f;
typedef __attribute__((ext_vector_type(8))) float v8f;
typedef __attribute__((ext_vector_type(4))) unsigned int v4u;
typedef __attribute__((ext_vector_type(8))) int v8i;
typedef __attribute__((ext_vector_type(4))) int v4i;

#define TOKENS_PER_BLOCK 64
#define KT    256                 // K-slab staged by TDM
#define LROW  (KT + 4)            // padded row stride in dwords (TDM pads 4 dwords / 256)
#define SLABF (64 * LROW)         // floats per slab  = 16640
#define SLABB (SLABF * 4)         // bytes per slab   = 66560
#define NEXP  64
#define HID   4096
#define TOPK  8
#define NSLAB (HID / KT)          // 16

// Issue a TDM DMA: weight[0:64, k0:k0+KT] (f32, row stride 4096) -> LDS at bufByteOff,
// with 4-dword padding appended after every 256-dword row (=> LROW dword rows).
__device__ __forceinline__ void tdm_load_weight_slab(const float* weight, int k0,
                                                     unsigned bufByteOff)
{
    const unsigned long long ga = (unsigned long long)(uintptr_t)(weight + k0);
    // D# group 0: count=1 | lds_addr | global_addr[56:0] | type=2
    v4u g0 = { 1u,
               bufByteOff,
               (unsigned)(ga & 0xFFFFFFFFu),
               (unsigned)((ga >> 32) & 0x01FFFFFFu) | (2u << 30) };
    // D# group 1:
    //  w0: data_size=2(4B) | pad_enable | pad_interval=7 (256 dw) | pad_amount=3 (4 dw)
    //  tensor_dim0=4096, tensor_dim1=64, tile_dim0=256, tile_dim1=64, dim0_stride=4096
    v8i g1 = { (int)((2u << 16) | (1u << 20) | (7u << 22) | (3u << 25)),
               (int)(((unsigned)HID & 0xFFFFu) << 16),   // tensor_dim0 low16 -> bits 63:48
               (int)(((unsigned)NEXP) << 16),            // tensor_dim1      -> bits 95:80
               (int)(((unsigned)KT) << 16),              // tile_dim0        -> bits 127:112
               (int)NEXP,                                // tile_dim1        -> bits 143:128
               (int)HID,                                 // tensor_dim0_stride -> bits 191:160
               0, 0 };
    v4i gz4 = { 0, 0, 0, 0 };                            // groups 2/3 unused (2-D tile)
    v8i gz8 = { 0, 0, 0, 0, 0, 0, 0, 0 };
    __builtin_amdgcn_tensor_load_to_lds(g0, g1, gz4, gz4, gz8, 0);
}

extern "C" __global__ __launch_bounds__(128)
void router_gemm_softmax_topk(const float* __restrict__ x,
                              const float* __restrict__ weight,
                              float* __restrict__ wOut,
                              int*   __restrict__ iOut)
{
    extern __shared__ float lds[];   // 2 slabs, 133120 B; slab0 reused for logits

    const int tid  = threadIdx.x;
    const int lane = tid & 31;
    const int wave = tid >> 5;
    const int blockTok = blockIdx.x * TOKENS_PER_BLOCK;
    const int waveTok  = blockTok + wave * 16;

    // 32-bit A 16x4 WMMA layout: lanes 0-15 hold K0,K1 ; lanes 16-31 hold K2,K3
    const int m    = lane & 15;
    const int koff = (lane >> 4) << 1;

    v8f acc0 = {}, acc1 = {}, acc2 = {}, acc3 = {};
    const float* aBase = x + (size_t)(waveTok + m) * HID + koff;

    // prime the pipeline: slab 0 -> buffer 0
    if (wave == 0) tdm_load_weight_slab(weight, 0, 0u);

    for (int j = 0; j < NSLAB; ++j) {
        if (wave == 0) __builtin_amdgcn_s_wait_tensorcnt(0);   // slab j landed in LDS
        __syncthreads();
        if (wave == 0 && (j + 1) < NSLAB)                      // prefetch next slab
            tdm_load_weight_slab(weight, (j + 1) * KT,
                                 (unsigned)(((j + 1) & 1) * SLABB));

        const float* buf   = lds + (j & 1) * SLABF;
        const float* bBase = buf + (lane & 15) * LROW + koff;
        const float* aK    = aBase + j * KT;

        #pragma unroll 4
        for (int kk = 0; kk < KT; kk += 4) {
            const v2f a  = *(const v2f*)(aK + kk);
            const v2f b0 = *(const v2f*)(bBase + kk);
            const v2f b1 = *(const v2f*)(bBase + 16 * LROW + kk);
            const v2f b2 = *(const v2f*)(bBase + 32 * LROW + kk);
            const v2f b3 = *(const v2f*)(bBase + 48 * LROW + kk);
            acc0 = __builtin_amdgcn_wmma_f32_16x16x4_f32(false, a, false, b0, (short)0, acc0, false, false);
            acc1 = __builtin_amdgcn_wmma_f32_16x16x4_f32(false, a, false, b1, (short)0, acc1, false, false);
            acc2 = __builtin_amdgcn_wmma_f32_16x16x4_f32(false, a, false, b2, (short)0, acc2, false, false);
            acc3 = __builtin_amdgcn_wmma_f32_16x16x4_f32(false, a, false, b3, (short)0, acc3, false, false);
        }
        __syncthreads();   // all reads of buf(j&1) done before it is overwritten at j+2
    }

    // ---- scatter logits to LDS as [64 tokens][65] (padded rows, slab0 area) ----
    // C/D layout: lane l holds column N=(l&15), rows M = v + 8*(l>=16)
    const int tokLoc = wave * 16 + ((lane >> 4) << 3);
    const int col    = lane & 15;
    #pragma unroll
    for (int v = 0; v < 8; ++v) {
        lds[(tokLoc + v) * 65 + col]      = acc0[v];
        lds[(tokLoc + v) * 65 + col + 16] = acc1[v];
        lds[(tokLoc + v) * 65 + col + 32] = acc2[v];
        lds[(tokLoc + v) * 65 + col + 48] = acc3[v];
    }
    __syncthreads();

    // ---- per-token softmax + top-8 (one thread per token) ----
    if (tid < TOKENS_PER_BLOCK) {
        const float* row = lds + tid * 65;
        float mx = row[0];
        #pragma unroll
        for (int j = 1; j < NEXP; ++j) mx = fmaxf(mx, row[j]);
        float sum = 0.f;
        #pragma unroll
        for (int j = 0; j < NEXP; ++j) sum += __expf(row[j] - mx);
        const float inv = 1.f / sum;

        unsigned long long used = 0ull;
        const int tokG = blockTok + tid;
        #pragma unroll
        for (int s = 0; s < TOPK; ++s) {
            float best = -INFINITY; int bi = 0;
            for (int j = 0; j < NEXP; ++j) {
                if (used & (1ull << j)) continue;
                const float vv = row[j];
                if (vv > best) { best = vv; bi = j; }   // strict > : lowest index wins ties
            }
            used |= (1ull << bi);
            wOut[tokG * TOPK + s] = __expf(best - mx) * inv;
            iOut[tokG * TOPK + s] = bi;
        }
    }
}

extern "C" void kernel_launch(void* const* d_in, const int* in_sizes, int n_in,
                              void* d_out, int out_size, void* d_ws, size_t ws_size,
                              hipStream_t stream) {
    const float* x = (const float*)d_in[0];   // [16384, 4096] f32
    const float* w = (const float*)d_in[1];   // [64, 4096] f32
    float* wOut = (float*)d_out;                       // [16384, 8] f32
    int*   iOut = (int*)d_out + (out_size / 2);        // [16384, 8] int32 (raw bits)

    const int nTokens = in_sizes[0] / HID;             // 16384
    const size_t shmem = 2u * SLABB;                   // 133120 B (double-buffered)

    dim3 grid(nTokens / TOKENS_PER_BLOCK), block(128); // 256 blocks x 4 waves
    router_gemm_softmax_topk<<<grid, block, shmem, stream>>>(x, w, wOut, iOut);
}